// GCNNetwork_1082331758969
// MI455X (gfx1250) — compile-verified
//
#include <hip/hip_runtime.h>

#define NNODES 4096
#define FEAT 200
#define GEMM_WAVES 8          // waves per block sharing one B slab
#define BSTRIDE 202           // LDS row stride (even for b64 align; 202%64=10 decorrelates banks)

typedef __attribute__((ext_vector_type(2))) float v2f;
typedef __attribute__((ext_vector_type(8))) float v8f;

// ---------------------------------------------------------------- utilities
__global__ void k_fill(float* __restrict__ p, float v, int n) {
  int i = blockIdx.x * blockDim.x + threadIdx.x;
  if (i < n) p[i] = v;
}

__global__ void k_deg_scatter(const int* __restrict__ col, const float* __restrict__ w,
                              float* __restrict__ deg, int e_cnt) {
  int i = blockIdx.x * blockDim.x + threadIdx.x;
  if (i < e_cnt) atomicAdd(&deg[col[i]], w[i]);
}

__global__ void k_rsqrt(float* __restrict__ p, int n) {
  int i = blockIdx.x * blockDim.x + threadIdx.x;
  if (i < n) p[i] = rsqrtf(p[i]);
}

__global__ void k_row_scatter_total(const int* __restrict__ row, const float* __restrict__ w,
                                    float* __restrict__ rowsum, float* __restrict__ total,
                                    int e_cnt) {
  __shared__ float red[256];
  int i = blockIdx.x * blockDim.x + threadIdx.x;
  float v = 0.f;
  if (i < e_cnt) { v = w[i]; atomicAdd(&rowsum[row[i]], v); }
  red[threadIdx.x] = v;
  __syncthreads();
  for (int s = 128; s > 0; s >>= 1) {
    if ((int)threadIdx.x < s) red[threadIdx.x] += red[threadIdx.x + s];
    __syncthreads();
  }
  if (threadIdx.x == 0) atomicAdd(total, red[0]);
}

__global__ void k_feat_div(const float* __restrict__ rowsum, const float* __restrict__ total,
                           float* __restrict__ feat, int n) {
  int i = blockIdx.x * blockDim.x + threadIdx.x;
  if (i < n) feat[i] = rowsum[i] / total[0];
}

// h[i][j] = feat[i] * W[j]  (GCN with in-dim 1)
__global__ void k_outer1(const float* __restrict__ feat, const float* __restrict__ w200,
                         float* __restrict__ h, int n) {
  int i = blockIdx.x * blockDim.x + threadIdx.x;
  if (i < n) h[i] = feat[i / FEAT] * w200[i % FEAT];
}

// self-loop contribution: out = h * dinv^2
__global__ void k_selfloop(const float* __restrict__ h, const float* __restrict__ dinv,
                           float* __restrict__ out, int n) {
  int i = blockIdx.x * blockDim.x + threadIdx.x;
  if (i < n) { float d = dinv[i / FEAT]; out[i] = h[i] * d * d; }
}

// one wave32 per edge: out[col] += h[row] * (dinv[row]*w*dinv[col])
__global__ void k_aggregate(const float* __restrict__ h, const int* __restrict__ row,
                            const int* __restrict__ col, const float* __restrict__ w,
                            const float* __restrict__ dinv, float* __restrict__ out, int e_cnt) {
  int gid = blockIdx.x * blockDim.x + threadIdx.x;
  int e = gid >> 5;
  int lane = gid & 31;
  if (e >= e_cnt) return;
  int r = row[e], c = col[e];
  float norm = dinv[r] * w[e] * dinv[c];
  const float* hs = h + (size_t)r * FEAT;
  float* od = out + (size_t)c * FEAT;
  __builtin_prefetch(hs + lane, 0, 0);  // global_prefetch_b8
  for (int j = lane; j < FEAT; j += 32)
    atomicAdd(&od[j], hs[j] * norm);
}

__global__ void k_bias_act(float* __restrict__ x, const float* __restrict__ b, int n, int relu) {
  int i = blockIdx.x * blockDim.x + threadIdx.x;
  if (i < n) {
    float v = x[i] + b[i % FEAT];
    if (relu) v = fmaxf(v, 0.f);
    x[i] = v;
  }
}

// -------------------------------------------------- FP32 WMMA GEMM (16x16x4)
// C = A[MxK] * B[KxN] (+bias)(+relu). K % 4 == 0. 256 threads = 8 waves per
// block; the block stages one zero-padded 16-column slab of B into LDS
// (transposed, [col][k]) and each wave computes one 16x16 tile against it.
// Inner loop is branch-free: one global b64 (A pair) + one ds b64 (B pair)
// per v_wmma_f32_16x16x4_f32.
__global__ __launch_bounds__(256) void k_wmma_gemm(
    const float* __restrict__ A, const float* __restrict__ B,
    const float* __restrict__ bias, float* __restrict__ C,
    int M, int K, int Ncols, int relu) {
  __shared__ float ldsBt[16 * BSTRIDE];

  const int tid = threadIdx.x;
  const int n0 = blockIdx.y * 16;

  // stage B slab: coalesced global reads (16 consecutive cols per row),
  // zero-pad columns beyond Ncols so the compute loop needs no guards.
  for (int idx = tid; idx < 16 * K; idx += 256) {
    int kk = idx >> 4;
    int cc = idx & 15;
    int n = n0 + cc;
    float v = (n < Ncols) ? B[(size_t)kk * Ncols + n] : 0.f;
    ldsBt[cc * BSTRIDE + kk] = v;
  }
  __syncthreads();

  const int wave = tid >> 5;
  const int lane = tid & 31;
  const int half = lane >> 4;      // 0: K+{0,1}, 1: K+{2,3}
  const int l = lane & 15;
  const int m0 = (blockIdx.x * GEMM_WAVES + wave) * 16;
  const int n = n0 + l;

  v8f acc = {};
  const float* arow = A + (size_t)(m0 + l) * K + 2 * half;   // 8B-aligned (K even)
  const float* brow = &ldsBt[l * BSTRIDE + 2 * half];        // 8B-aligned (BSTRIDE even)
  for (int k = 0; k < K; k += 4) {
    v2f af = *(const v2f*)(arow + k);
    v2f bf = *(const v2f*)(brow + k);
    acc = __builtin_amdgcn_wmma_f32_16x16x4_f32(false, af, false, bf, (short)0, acc,
                                                false, false);
  }
  if (n < Ncols) {
    for (int r = 0; r < 8; ++r) {
      float v = acc[r];
      if (bias) v += bias[n];
      if (relu) v = fmaxf(v, 0.f);
      C[(size_t)(m0 + r + 8 * half) * Ncols + n] = v;
    }
  }
}

// ---------------------------------------------------------------- small tails
__global__ void k_col_mean(const float* __restrict__ h, float* __restrict__ out, int M, int Ncols) {
  int j = blockIdx.x * blockDim.x + threadIdx.x;
  if (j >= Ncols) return;
  float s = 0.f;
  for (int i = 0; i < M; ++i) s += h[(size_t)i * Ncols + j];
  out[j] = s * (1.0f / (float)M);
}

__global__ void k_routing(const float* __restrict__ rin_g,
                          const float* __restrict__ W0, const float* __restrict__ b0,
                          const float* __restrict__ W1, const float* __restrict__ b1,
                          const float* __restrict__ W2, const float* __restrict__ b2,
                          float* __restrict__ out) {
  __shared__ float rin[100], s1[50], s2[50];
  int t = threadIdx.x;
  if (t < 100) rin[t] = rin_g[t];
  __syncthreads();
  if (t < 50) {
    float a = b0[t];
    for (int i = 0; i < 100; ++i) a += rin[i] * W0[i * 50 + t];
    s1[t] = fmaxf(a, 0.f);
  }
  __syncthreads();
  if (t < 50) {
    float a = b1[t];
    for (int i = 0; i < 50; ++i) a += s1[i] * W1[i * 50 + t];
    s2[t] = fmaxf(a, 0.f);
  }
  __syncthreads();
  if (t < 200) {
    float a = b2[t];
    for (int i = 0; i < 50; ++i) a += s2[i] * W2[i * 200 + t];
    out[t] = a;
  }
}

__global__ void k_dense(const float* __restrict__ x, const float* __restrict__ W,
                        const float* __restrict__ b, float* __restrict__ out,
                        int in_dim, int out_dim, int relu) {
  int j = blockIdx.x * blockDim.x + threadIdx.x;
  if (j >= out_dim) return;
  float a = b[j];
  for (int i = 0; i < in_dim; ++i) a += x[i] * W[(size_t)i * out_dim + j];
  if (relu) a = fmaxf(a, 0.f);
  out[j] = a;
}

// ---------------------------------------------------------------- launch
extern "C" void kernel_launch(void* const* d_in, const int* in_sizes, int n_in,
                              void* d_out, int out_size, void* d_ws, size_t ws_size,
                              hipStream_t stream) {
  (void)out_size; (void)ws_size;
  const int N = NNODES, F = FEAT;
  const int* topo_ei   = (const int*)d_in[0];
  const float* topo_w  = (const float*)d_in[1];
  const int* traf_ei   = (const int*)d_in[2];
  const float* traf_w  = (const float*)d_in[3];
  const float* routing = (const float*)d_in[4];
  const int E_cnt = in_sizes[1];

  const float* P[38];
  for (int i = 0; i < 38 && (5 + i) < n_in; ++i) P[i] = (const float*)d_in[5 + i];
  const float *tg0W=P[0],  *tg0b=P[1],  *tg1W=P[2],  *tg1b=P[3];
  const float *tg2W=P[4],  *tg2b=P[5],  *tg3W=P[6],  *tg3b=P[7];
  const float *tl0W=P[8],  *tl0b=P[9],  *tl1W=P[10], *tl1b=P[11], *tl2W=P[12], *tl2b=P[13];
  const float *fg0W=P[14], *fg0b=P[15], *fg1W=P[16], *fg1b=P[17];
  const float *fg2W=P[18], *fg2b=P[19], *fg3W=P[20], *fg3b=P[21];
  const float *fl0W=P[22], *fl0b=P[23], *fl1W=P[24], *fl1b=P[25], *fl2W=P[26], *fl2b=P[27];
  const float *r0W=P[28], *r0b=P[29], *r1W=P[30], *r1b=P[31], *r2W=P[32], *r2b=P[33];
  const float *j0W=P[34], *j0b=P[35], *j1W=P[36], *j1b=P[37];

  float* ws     = (float*)d_ws;
  float* bufA   = ws;                       // N*F
  float* bufB   = bufA + (size_t)N * F;     // N*F
  float* dinvT  = bufB + (size_t)N * F;     // N
  float* dinvF  = dinvT + N;                // N
  float* rowsum = dinvF + N;                // N
  float* feat   = rowsum + N;               // N
  float* total  = feat + N;                 // 8
  float* cat    = total + 8;                // 600 (+pad)
  float* hj     = cat + 608;                // 1024

  const int NF = N * F;
  dim3 bs(256);
  int gNF = (NF + 255) / 256;
  int gE  = (E_cnt + 255) / 256;
  int gEw = (E_cnt * 32 + 255) / 256;       // one wave per edge
  int gN  = (N + 255) / 256;
  dim3 gemmGrid((unsigned)(N / (16 * GEMM_WAVES)), (unsigned)((F + 15) / 16));

  // symmetric-normalization coefficients (deg includes self-loop weight 1)
  k_fill<<<gN, bs, 0, stream>>>(dinvT, 1.0f, N);
  k_deg_scatter<<<gE, bs, 0, stream>>>(topo_ei + E_cnt, topo_w, dinvT, E_cnt);
  k_rsqrt<<<gN, bs, 0, stream>>>(dinvT, N);
  k_fill<<<gN, bs, 0, stream>>>(dinvF, 1.0f, N);
  k_deg_scatter<<<gE, bs, 0, stream>>>(traf_ei + E_cnt, traf_w, dinvF, E_cnt);
  k_rsqrt<<<gN, bs, 0, stream>>>(dinvF, N);

  // topology node feature = row-sums / total
  k_fill<<<gN, bs, 0, stream>>>(rowsum, 0.0f, N);
  k_fill<<<1, bs, 0, stream>>>(total, 0.0f, 8);
  k_row_scatter_total<<<gE, bs, 0, stream>>>(topo_ei, topo_w, rowsum, total, E_cnt);
  k_feat_div<<<gN, bs, 0, stream>>>(rowsum, total, feat, N);

  auto agg = [&](const float* h, float* out, const int* ei, const float* w,
                 const float* dinv, const float* b) {
    k_selfloop<<<gNF, bs, 0, stream>>>(h, dinv, out, NF);
    k_aggregate<<<gEw, bs, 0, stream>>>(h, ei, ei + E_cnt, w, dinv, out, E_cnt);
    k_bias_act<<<gNF, bs, 0, stream>>>(out, b, NF, 1);
  };
  auto gemm = [&](const float* A, const float* W, const float* b, float* C, int relu) {
    k_wmma_gemm<<<gemmGrid, bs, 0, stream>>>(A, W, b, C, N, F, F, relu);
  };

  // -------- topology branch --------
  k_outer1<<<gNF, bs, 0, stream>>>(feat, tg0W, bufA, NF);       // h0 = feat ⊗ W
  agg(bufA, bufB, topo_ei, topo_w, dinvT, tg0b);
  gemm(bufB, tg1W, nullptr, bufA, 0);
  agg(bufA, bufB, topo_ei, topo_w, dinvT, tg1b);
  gemm(bufB, tl0W, tl0b, bufA, 1);
  gemm(bufA, tg2W, nullptr, bufB, 0);
  agg(bufB, bufA, topo_ei, topo_w, dinvT, tg2b);
  gemm(bufA, tg3W, nullptr, bufB, 0);
  agg(bufB, bufA, topo_ei, topo_w, dinvT, tg3b);
  gemm(bufA, tl1W, tl1b, bufB, 1);
  gemm(bufB, tl2W, tl2b, bufA, 0);
  k_col_mean<<<1, bs, 0, stream>>>(bufA, cat + 0, N, F);

  // -------- traffic branch (x = I  =>  h0 = fg0W directly) --------
  agg(fg0W, bufB, traf_ei, traf_w, dinvF, fg0b);
  gemm(bufB, fg1W, nullptr, bufA, 0);
  agg(bufA, bufB, traf_ei, traf_w, dinvF, fg1b);
  gemm(bufB, fl0W, fl0b, bufA, 1);
  gemm(bufA, fg2W, nullptr, bufB, 0);
  agg(bufB, bufA, traf_ei, traf_w, dinvF, fg2b);
  gemm(bufA, fg3W, nullptr, bufB, 0);
  agg(bufB, bufA, traf_ei, traf_w, dinvF, fg3b);
  gemm(bufA, fl1W, fl1b, bufB, 1);
  gemm(bufB, fl2W, fl2b, bufA, 0);
  k_col_mean<<<1, bs, 0, stream>>>(bufA, cat + 200, N, F);

  // -------- routing MLP --------
  k_routing<<<1, bs, 0, stream>>>(routing, r0W, r0b, r1W, r1b, r2W, r2b, cat + 400);

  // -------- joint head --------
  k_dense<<<4, bs, 0, stream>>>(cat, j0W, j0b, hj, 600, 1024, 1);
  k_dense<<<2, bs, 0, stream>>>(hj, j1W, j1b, (float*)d_out, 1024, 512, 0);
}